// CTC_15333033246839
// MI455X (gfx1250) — compile-verified
//
#include <hip/hip_runtime.h>
#include <stdint.h>

#define TT 512
#define BB 32
#define CC 6000
#define LL 32
#define SS 65          // 2L+1
#define SP 68          // padded state stride (16B aligned)
#define NEGV (-1e30f)

typedef float v2f __attribute__((ext_vector_type(2)));
typedef float v8f __attribute__((ext_vector_type(8)));
typedef int   v4i __attribute__((ext_vector_type(4)));

#if __has_builtin(__builtin_amdgcn_global_load_async_to_lds_b128)
#define ASYNC_OK 1
typedef __attribute__((address_space(1))) v4i* as1v4;
typedef __attribute__((address_space(3))) v4i* as3v4;
#endif

__device__ __forceinline__ float lse3f(float a, float b, float c) {
  float m = fmaxf(a, fmaxf(b, c));
  float r = __expf(a - m) + __expf(b - m) + __expf(c - m);
  return m + __logf(r);
}

// ---------------- Kernel 0: build extended labels / skip masks / lengths ---
__global__ void k_setup(const int* __restrict__ label, int* __restrict__ ext,
                        int* __restrict__ skip, int* __restrict__ len) {
  int tid = threadIdx.x;
  if (tid < BB) {
    int c = 0;
    for (int l = 0; l < LL; ++l) c += (label[tid * LL + l] >= 0) ? 1 : 0;
    len[tid] = c;
  }
  for (int i = tid; i < BB * SS; i += blockDim.x) {
    int b = i / SS, s = i % SS;
    int e = 0;
    if (s & 1) { int lab = label[b * LL + (s >> 1)]; e = (lab >= 0) ? lab + 1 : 0; }
    ext[i] = e;
    int sk = 0;
    if ((s & 1) && s >= 2) {
      int lab2 = label[b * LL + ((s - 2) >> 1)];
      int e2 = (lab2 >= 0) ? lab2 + 1 : 0;
      sk = (e != e2) ? 1 : 0;
    }
    skip[i] = sk;
  }
}

// ---------------- Kernel 1: per-(t,b) row logsumexp + label gather ---------
// one block (256 thr = 8 waves) per row; single pass over global memory.
__global__ void __launch_bounds__(256) k_row(const float* __restrict__ in,
                                             const int* __restrict__ ext,
                                             float* __restrict__ lp) {
  int r = blockIdx.x;                 // r = t*B + b
  int t = r / BB, b = r % BB;
  const float* p = in + (size_t)r * CC;
  const float4* p4 = (const float4*)p;          // 1500 float4 per row
  int tid = threadIdx.x, wave = tid >> 5, lane = tid & 31;

  float4 v[6];
  #pragma unroll
  for (int k = 0; k < 6; ++k) {
    int i = tid + k * 256;
    v[k] = (i < 1500) ? p4[i] : make_float4(NEGV, NEGV, NEGV, NEGV);
  }
  float m = NEGV;
  #pragma unroll
  for (int k = 0; k < 6; ++k)
    m = fmaxf(m, fmaxf(fmaxf(v[k].x, v[k].y), fmaxf(v[k].z, v[k].w)));
  #pragma unroll
  for (int off = 16; off; off >>= 1) m = fmaxf(m, __shfl_xor(m, off, 32));

  __shared__ float red[8];
  __shared__ float bc;
  if (lane == 0) red[wave] = m;
  __syncthreads();
  float rm = red[0];
  #pragma unroll
  for (int w = 1; w < 8; ++w) rm = fmaxf(rm, red[w]);
  __syncthreads();

  float s = 0.f;
  #pragma unroll
  for (int k = 0; k < 6; ++k) {
    s += __expf(v[k].x - rm) + __expf(v[k].y - rm) +
         __expf(v[k].z - rm) + __expf(v[k].w - rm);   // invalid slots -> exp(-1e30)=0
  }

  // wave-level sum via f32 WMMA against a ones matrix: D[m,n] = p_m + p_{m+16}
  float swave;
#if __has_builtin(__builtin_amdgcn_wmma_f32_16x16x4_f32)
  {
    v2f a; a[0] = s; a[1] = 0.f;
    v2f ones; ones[0] = 1.f; ones[1] = 1.f;
    v8f acc = {};
    acc = __builtin_amdgcn_wmma_f32_16x16x4_f32(false, a, false, ones,
                                                (short)0, acc, false, false);
    float q = acc[0] + acc[1] + acc[2] + acc[3] + acc[4] + acc[5] + acc[6] + acc[7];
    q += __shfl_xor(q, 16, 32);   // combine the two M-halves -> full 32-lane sum
    swave = q;
  }
#else
  {
    float q = s;
    #pragma unroll
    for (int off = 16; off; off >>= 1) q += __shfl_xor(q, off, 32);
    swave = q;
  }
#endif

  if (lane == 0) red[wave] = swave;
  __syncthreads();
  if (tid == 0) {
    float tot = 0.f;
    #pragma unroll
    for (int w = 0; w < 8; ++w) tot += red[w];
    bc = rm + __logf(tot);
  }
  __syncthreads();
  float lse = bc;

  // gather log-probs at extended-label classes into (b, t, SP) table
  for (int sidx = tid; sidx < SS; sidx += 256) {
    int c = ext[b * SS + sidx];
    lp[((size_t)b * TT + t) * SP + sidx] = p[c] - lse;
  }
}

// ---------------- chunk staging helpers (async-to-LDS when available) ------
#define CHUNK 64
#define CHUNK_F4 ((CHUNK * SP) / 4)   // 1088 float4 per chunk

__device__ __forceinline__ void chunk_issue(const float* __restrict__ g,
                                            float* __restrict__ l, int lane) {
#ifdef ASYNC_OK
  for (int i = lane; i < CHUNK_F4; i += 32) {
    as1v4 gsrc = (as1v4)(uintptr_t)(g + (size_t)i * 4);
    as3v4 ldst = (as3v4)(unsigned)(uintptr_t)(l + (size_t)i * 4);
    __builtin_amdgcn_global_load_async_to_lds_b128(gsrc, ldst, 0, 0);
  }
#else
  for (int i = lane; i < CHUNK_F4; i += 32)
    ((float4*)l)[i] = ((const float4*)g)[i];
#endif
}

__device__ __forceinline__ void chunk_wait() {
#ifdef ASYNC_OK
#if __has_builtin(__builtin_amdgcn_s_wait_asynccnt)
  __builtin_amdgcn_s_wait_asynccnt(0);
#else
  asm volatile("s_wait_asynccnt 0" ::: "memory");
#endif
#endif
}

// ---------------- Kernel 2: alpha recursion, one wave32 per batch ----------
__global__ void __launch_bounds__(32) k_alpha(const float* __restrict__ lp,
                                              const int* __restrict__ skip,
                                              const int* __restrict__ lenp,
                                              float* __restrict__ lossb) {
  __shared__ float buf0[CHUNK * SP];
  __shared__ float buf1[CHUNK * SP];
  int b = blockIdx.x, lane = threadIdx.x;
  const float* lpb = lp + (size_t)b * TT * SP;

  bool sk0 = skip[b * SS + lane] != 0;          // state s = lane
  bool sk1 = skip[b * SS + lane + 32] != 0;     // state s = lane+32 (<= 63)
  int len = lenp[b];

  chunk_issue(lpb, buf0, lane);
  chunk_wait();
  chunk_issue(lpb + (size_t)CHUNK * SP, buf1, lane);

  // t = 0 init
  float lp00 = buf0[0], lp01 = buf0[1];
  float A0 = (lane == 0) ? lp00 : ((lane == 1 && len > 0) ? lp01 : NEGV);
  float A1 = NEGV;
  float A2 = NEGV;   // state 64; uniform across lanes by construction

  for (int c = 0; c < 8; ++c) {
    const float* cur = (c & 1) ? buf1 : buf0;
    int tstart = (c == 0) ? 1 : 0;
    for (int tt = tstart; tt < CHUNK; ++tt) {
      const float* row = cur + tt * SP;
      float l0v = row[lane];
      float l1v = row[lane + 32];
      float l2v = row[64];                      // uniform broadcast
      float a31 = __shfl(A0, 31, 32);
      float a30 = __shfl(A0, 30, 32);
      float b63 = __shfl(A1, 31, 32);
      float b62 = __shfl(A1, 30, 32);
      (void)b62;
      float u01 = __shfl_up(A0, 1, 32); if (lane == 0) u01 = NEGV;
      float u02 = __shfl_up(A0, 2, 32); if (lane < 2) u02 = NEGV;
      float u11 = __shfl_up(A1, 1, 32); if (lane == 0) u11 = a31;
      float u12 = __shfl_up(A1, 2, 32);
      if (lane == 0) u12 = a30; else if (lane == 1) u12 = a31;
      float n0 = lse3f(A0, u01, sk0 ? u02 : NEGV) + l0v;
      float n1 = lse3f(A1, u11, sk1 ? u12 : NEGV) + l1v;
      float n2 = lse3f(A2, b63, NEGV) + l2v;    // state 64 is blank: no skip
      A0 = n0; A1 = n1; A2 = n2;
    }
    if (c < 7) {
      chunk_wait();                             // chunk c+1 landed
      if (c + 2 < 8)
        chunk_issue(lpb + (size_t)(c + 2) * CHUNK * SP,
                    (c & 1) ? buf1 : buf0, lane);
    }
  }

  int send = 2 * len;                           // <= 64
  float l1 = (send == 64) ? A2
           : (send >= 32) ? __shfl(A1, send - 32, 32)
                          : __shfl(A0, send, 32);
  int sm1 = (send - 1 > 0) ? send - 1 : 0;
  float l2 = (sm1 >= 32) ? __shfl(A1, sm1 - 32, 32) : __shfl(A0, sm1, 32);
  if (len == 0) l2 = NEGV;
  float mm = fmaxf(l1, l2);
  float loss = -(mm + __logf(__expf(l1 - mm) + __expf(l2 - mm)));
  if (!(loss < 1e29f)) loss = 0.f;
  if (lane == 0) lossb[b] = loss;
}

// ---------------- Kernel 3: final scalar --------------------------------
__global__ void k_final(const float* __restrict__ lossb, float* __restrict__ out) {
  float v = (threadIdx.x < BB) ? lossb[threadIdx.x] : 0.f;
  #pragma unroll
  for (int off = 16; off; off >>= 1) v += __shfl_xor(v, off, 32);
  if (threadIdx.x == 0) out[0] = v / (float)BB / (float)LL;
}

extern "C" void kernel_launch(void* const* d_in, const int* in_sizes, int n_in,
                              void* d_out, int out_size, void* d_ws, size_t ws_size,
                              hipStream_t stream) {
  (void)in_sizes; (void)n_in; (void)out_size; (void)ws_size;
  const float* in  = (const float*)d_in[0];
  const int* label = (const int*)d_in[1];
  float* out = (float*)d_out;
  char* ws = (char*)d_ws;
  int*   ext   = (int*)(ws);                 // B*S*4      = 8320 B
  int*   skip  = (int*)(ws + 8320);          // B*S*4      = 8320 B
  int*   len   = (int*)(ws + 16640);         // B*4        = 128 B
  float* lossb = (float*)(ws + 16768);       // B*4        = 128 B
  float* lp    = (float*)(ws + 32768);       // B*T*SP*4   = 4,456,448 B

  k_setup<<<1, 256, 0, stream>>>(label, ext, skip, len);
  k_row<<<TT * BB, 256, 0, stream>>>(in, ext, lp);
  k_alpha<<<BB, 32, 0, stream>>>(lp, skip, len, lossb);
  k_final<<<1, 32, 0, stream>>>(lossb, out);
}